// DigitCapsules_12515534700781
// MI455X (gfx1250) — compile-verified
//
#include <hip/hip_runtime.h>

// CapsNet DigitCapsules, fully fused for MI455X (gfx1250, wave32, WMMA).
// One workgroup = 4 batch elements; conv-GEMM via v_wmma_f32_16x16x4_f32,
// prediction GEMM, and 3 dynamic-routing iterations all stay in the 320KB
// WGP LDS. x read once from HBM via async-to-LDS DMA; zero workspace.

typedef float v2f __attribute__((ext_vector_type(2)));
typedef float v8f __attribute__((ext_vector_type(8)));

#define NUM_CAPS 10
#define OUT_CH   16
#define IN_CH    256
#define HW       36          // 6x6
#define PRIM_DIM 8
#define NUM_PRIM 72
#define F_TOT    576         // OUT_CH*HW = NUM_PRIM*PRIM_DIM
#define E_DIM    16
#define BT       4           // batches per workgroup
#define ROWS     (BT*HW)     // 144
#define MTILES   9           // 144/16
#define THREADS  256
#define WAVES    8

// ---- LDS layout (float offsets), phased aliasing ----
// x at 0 so phase-1 A-operand DS offsets fit the 16-bit immediate field.
// x     [BT][256][36]        : [0, 36864)            (phase1 only)
// cw    [16][260] padded     : [36864, 41024)        (phase1 only)
// u     [BT][10][72][16]     : [0, 46080)            (phase2 write, phase3 read; aliases x+cw)
// caps  [BT][10][576]        : [46080, 69120)        (phase1 write, phase2 read, phase3 scratch)
// w     [72][8][16]          : [69120, 78336)        (phase2 only)
#define X_OFF    0
#define CW_OFF   36864
#define U_OFF    0
#define CAPS_OFF 46080
#define W_OFF    69120
#define SMEM_FLOATS 78336    // 313,344 bytes < 320KB WGP LDS

// ---- CDNA5 async global->LDS DMA (ASYNCcnt-tracked, bypasses VGPRs) ----
__device__ __forceinline__ void async_b128_to_lds(const void* g, unsigned lds_byte) {
    unsigned long long ga = (unsigned long long)g;
    asm volatile("global_load_async_to_lds_b128 %0, %1, off"
                 :: "v"(lds_byte), "v"(ga) : "memory");
}
__device__ __forceinline__ void wait_async0() {
    asm volatile("s_wait_asynccnt 0x0" ::: "memory");
}

extern "C" __global__ __launch_bounds__(THREADS)
void capsnet_digitcaps_fused(const float* __restrict__ x,
                             const float* __restrict__ conv_w,
                             const float* __restrict__ conv_b,
                             const float* __restrict__ wts,
                             float* __restrict__ out)
{
    extern __shared__ float lds[];
    const int t    = threadIdx.x;
    const int lane = t & 31;
    const int wave = t >> 5;
    const int b0   = blockIdx.x * BT;

    // ---------- Phase 0: async-stage x tile (contiguous 144KB DMA) ----------
    {
        const char* gsrc = (const char*)(x + (size_t)b0 * (IN_CH * HW));
        #pragma unroll
        for (int i = 0; i < (BT * IN_CH * HW / 4) / THREADS; ++i) { // 36
            const unsigned off = (unsigned)(t + THREADS * i) * 16u;
            async_b128_to_lds(gsrc + off, (unsigned)(X_OFF * 4) + off);
        }
    }
    wait_async0();
    __syncthreads();

    // WMMA f32 16x16x4 lane decomposition:
    //  A: lane holds A[m = lane&15][k2], A[m][k2+1], k2 = (lane>>4)*2
    //  B: lane holds B[k2][n = lane&15], B[k2+1][n]
    //  D: vgpr i holds rows i (lanes 0-15) / i+8 (lanes 16-31), col n = lane&15
    const int n_lane = lane & 15;
    const int k2     = (lane >> 4) * 2;
    const int half   = (lane >> 4) * 8;

    // ---------- Phase 1: GEMM1  caps[b][k][o*36+s] = x·conv_w^T + bias ----------
    for (int k = 0; k < NUM_CAPS; ++k) {
        // async-stage conv_w[k] into padded [o][260] rows (row-interior is
        // contiguous so b128 DMA chunks land correctly)
        {
            const char* gcw = (const char*)(conv_w + (size_t)k * (OUT_CH * IN_CH));
            #pragma unroll
            for (int i = 0; i < (OUT_CH * IN_CH / 4) / THREADS; ++i) { // 4
                const int chunk = t + THREADS * i;          // < 1024
                const int o = chunk >> 6, cq = chunk & 63;  // 64 b128 per row
                async_b128_to_lds(gcw + (size_t)chunk * 16,
                                  (unsigned)((CW_OFF + o * 260 + cq * 4) * 4));
            }
        }
        wait_async0();
        __syncthreads();

        const float bias = conv_b[k * OUT_CH + n_lane];

        for (int mt = wave; mt < MTILES; mt += WAVES) {             // wave-uniform
            const int row = mt * 16 + n_lane;                       // A row (b_local*36+s)
            const int bl  = row / HW;
            const int s   = row - bl * HW;
            const float* aptr = lds + X_OFF + bl * (IN_CH * HW) + s; // + c*36 (imm-foldable)
            const float* bptr = lds + CW_OFF + n_lane * 260;         // + c    (imm-foldable)

            v8f acc = {0.f, 0.f, 0.f, 0.f, 0.f, 0.f, 0.f, 0.f};
            #pragma unroll
            for (int cs = 0; cs < IN_CH / 4; ++cs) {                // 64 WMMAs, fully unrolled
                const int c = cs * 4 + k2;
                v2f a; a.x = aptr[c * HW]; a.y = aptr[(c + 1) * HW];
                v2f b = *(const v2f*)(bptr + c);                    // ds_load_b64
                acc = __builtin_amdgcn_wmma_f32_16x16x4_f32(
                        false, a, false, b, (short)0, acc, false, false);
            }
            // scatter D tile into caps[b][k][o*36+s] (+bias)
            #pragma unroll
            for (int i = 0; i < 8; ++i) {
                const int r  = mt * 16 + half + i;
                const int rb = r / HW;
                const int rs = r - rb * HW;
                lds[CAPS_OFF + rb * (NUM_CAPS * F_TOT) + k * F_TOT
                    + n_lane * HW + rs] = acc[i] + bias;
            }
        }
        __syncthreads();
    }

    // ---------- Phase 2: GEMM2  u[b][k][p][e] = prim[b][k][p][:]·W[k][p] ----------
    for (int k = 0; k < NUM_CAPS; ++k) {
        // async-stage weights[k] raw ([p][d][e], pure b128 memcpy)
        {
            const char* gw = (const char*)(wts + (size_t)k * (NUM_PRIM * PRIM_DIM * E_DIM));
            #pragma unroll
            for (int i = 0; i < (NUM_PRIM * PRIM_DIM * E_DIM / 4) / THREADS; ++i) { // 9
                const int chunk = t + THREADS * i;          // < 2304
                async_b128_to_lds(gw + (size_t)chunk * 16,
                                  (unsigned)((W_OFF + chunk * 4) * 4));
            }
        }
        wait_async0();
        __syncthreads();

        #pragma unroll
        for (int i = 0; i < (BT * NUM_PRIM * E_DIM) / THREADS; ++i) {       // 18
            int idx = t + THREADS * i;
            int e = idx & 15;
            int p = (idx >> 4) % NUM_PRIM;
            int b = idx / (NUM_PRIM * E_DIM);
            const float* cp = lds + CAPS_OFF + b * (NUM_CAPS * F_TOT)
                              + k * F_TOT + p * PRIM_DIM;           // 8 contiguous (b128 x2)
            const float* wp = lds + W_OFF + p * (PRIM_DIM * E_DIM) + e; // + d*16
            float acc = 0.f;
            #pragma unroll
            for (int d = 0; d < PRIM_DIM; ++d) acc += cp[d] * wp[d * E_DIM];
            lds[U_OFF + b * (NUM_CAPS * NUM_PRIM * E_DIM)
                + k * (NUM_PRIM * E_DIM) + p * E_DIM + e] = acc;
        }
        __syncthreads();
    }

    // ---------- Phase 3: dynamic routing (3 iters), scratch reuses caps region ----
    float* bl_l = lds + CAPS_OFF;        // [BT][10][72] logits
    float* c_l  = lds + CAPS_OFF + 2880; // [BT][10][72] couplings
    float* s_l  = lds + CAPS_OFF + 5760; // [BT][10][16]
    float* v_l  = lds + CAPS_OFF + 6400; // [BT][10][16]
    float* nf_l = lds + CAPS_OFF + 7040; // [BT][10] squash factors

    for (int i = t; i < BT * NUM_CAPS * NUM_PRIM; i += THREADS) bl_l[i] = 0.f;
    __syncthreads();

    for (int it = 0; it < 3; ++it) {
        // c = softmax_k(b) per (batch, primary)
        for (int i = t; i < BT * NUM_PRIM; i += THREADS) {          // 288
            int b = i / NUM_PRIM, p = i - b * NUM_PRIM;
            const float* bp = bl_l + b * (NUM_CAPS * NUM_PRIM) + p;
            float m = bp[0];
            #pragma unroll
            for (int kk = 1; kk < NUM_CAPS; ++kk) m = fmaxf(m, bp[kk * NUM_PRIM]);
            float ex[NUM_CAPS]; float sum = 0.f;
            #pragma unroll
            for (int kk = 0; kk < NUM_CAPS; ++kk) {
                ex[kk] = __expf(bp[kk * NUM_PRIM] - m);
                sum += ex[kk];
            }
            const float inv = 1.0f / sum;
            float* cp = c_l + b * (NUM_CAPS * NUM_PRIM) + p;
            #pragma unroll
            for (int kk = 0; kk < NUM_CAPS; ++kk) cp[kk * NUM_PRIM] = ex[kk] * inv;
        }
        __syncthreads();

        // s[b][k][e] = sum_p c[b][k][p] * u[b][k][p][e]
        for (int i = t; i < BT * NUM_CAPS * E_DIM; i += THREADS) {  // 640
            int e = i & 15, kk = (i >> 4) % NUM_CAPS, b = i / (NUM_CAPS * E_DIM);
            const float* cc = c_l + b * (NUM_CAPS * NUM_PRIM) + kk * NUM_PRIM;
            const float* uu = lds + U_OFF + b * (NUM_CAPS * NUM_PRIM * E_DIM)
                              + kk * (NUM_PRIM * E_DIM) + e;
            float acc = 0.f;
            for (int p = 0; p < NUM_PRIM; ++p) acc += cc[p] * uu[p * E_DIM];
            s_l[i] = acc;
        }
        __syncthreads();

        // squash scale per (b, k)
        if (t < BT * NUM_CAPS) {
            const float* sp = s_l + t * E_DIM;
            float sq = 0.f;
            #pragma unroll
            for (int e = 0; e < E_DIM; ++e) sq += sp[e] * sp[e];
            nf_l[t] = (sq / (1.0f + sq)) * __frsqrt_rn(sq + 1e-8f);
        }
        __syncthreads();
        for (int i = t; i < BT * NUM_CAPS * E_DIM; i += THREADS)
            v_l[i] = s_l[i] * nf_l[i >> 4];
        __syncthreads();

        // b += u·v (skip on final iteration; v is the output)
        if (it < 2) {
            for (int i = t; i < BT * NUM_CAPS * NUM_PRIM; i += THREADS) { // 2880
                int p = i % NUM_PRIM, kk = (i / NUM_PRIM) % NUM_CAPS;
                int b = i / (NUM_CAPS * NUM_PRIM);
                const float4* uu = (const float4*)(lds + U_OFF
                    + b * (NUM_CAPS * NUM_PRIM * E_DIM)
                    + kk * (NUM_PRIM * E_DIM) + p * E_DIM);
                const float4* vv = (const float4*)(v_l + (b * NUM_CAPS + kk) * E_DIM);
                float acc = 0.f;
                #pragma unroll
                for (int q = 0; q < 4; ++q) {
                    float4 u4 = uu[q], v4 = vv[q];
                    acc += u4.x * v4.x + u4.y * v4.y + u4.z * v4.z + u4.w * v4.w;
                }
                bl_l[i] += acc;
            }
            __syncthreads();
        }
    }

    // ---------- write v: [B][10][16], coalesced ----------
    for (int i = t; i < BT * NUM_CAPS * E_DIM; i += THREADS)
        out[(size_t)b0 * (NUM_CAPS * E_DIM) + i] = v_l[i];
}

extern "C" void kernel_launch(void* const* d_in, const int* in_sizes, int n_in,
                              void* d_out, int out_size, void* d_ws, size_t ws_size,
                              hipStream_t stream) {
    (void)n_in; (void)out_size; (void)d_ws; (void)ws_size;
    const float* x      = (const float*)d_in[0];
    const float* conv_w = (const float*)d_in[1];
    const float* conv_b = (const float*)d_in[2];
    const float* wts    = (const float*)d_in[3];
    float* out = (float*)d_out;

    const int B = in_sizes[0] / (IN_CH * HW);       // 2048
    const size_t smem = (size_t)SMEM_FLOATS * sizeof(float); // 313,344 B

    hipFuncSetAttribute(reinterpret_cast<const void*>(capsnet_digitcaps_fused),
                        hipFuncAttributeMaxDynamicSharedMemorySize, (int)smem);
    capsnet_digitcaps_fused<<<B / BT, THREADS, smem, stream>>>(
        x, conv_w, conv_b, wts, out);
}